// SingleRoIExtractor_51951924413003
// MI455X (gfx1250) — compile-verified
//
#include <hip/hip_runtime.h>
#include <stdint.h>

// SingleRoIExtractor for MI455X (gfx1250).
// Memory-bound gather op: all 4 FPN levels (165MB) fit in the 192MB L2, so the
// kernel is L2-gather bound (~216MB HBM steady state => ~9us floor @ 23.3TB/s).
// WMMA is intentionally NOT used: bilinear interp is a K_eff=4 sparse contraction;
// dense 16x16x4 tiles would issue ~100x excess flops. CDNA5 paths used instead:
// async global->LDS staging of the RoI table (ASYNCcnt + s_wait_asynccnt),
// global prefetch hints, wave32 blocks with 16 independent gathers in flight.

#define OUTS   7
#define NBINS  49     // 7*7
#define CCH    256
#define SRK    2

struct RoiRec {
    float x1, y1, bw, bh;   // box origin + per-bin step, at level scale, aligned (-0.5)
    int   lvl, b, pad0, pad1;
};

// ---------------------------------------------------------------------------
// Kernel 1: per-RoI setup. Stages the RoI table into LDS via the gfx1250
// async-copy path, then computes level assignment + aligned box geometry.
// ---------------------------------------------------------------------------
__global__ __launch_bounds__(256)
void setup_rois_kernel(const float* __restrict__ rois, RoiRec* __restrict__ recs, int K)
{
    __shared__ float sro[256 * 5];          // only LDS object -> starts at LDS offset 0

    const int tid  = threadIdx.x;
    const int base = blockIdx.x * 256;      // this block handles rois [base, base+256)
    const float* gsrc = rois + (size_t)base * 5;
    const int nflt = min(K - base, 256) * 5;

    // Async global -> LDS copy (CDNA5 GLOBAL_LOAD_ASYNC_TO_LDS_B32, ASYNCcnt).
    // NOTE: &sro[i] is passed as an extra asm operand so the shared array's
    // address escapes -- otherwise GlobalOpt sees a store-free internal global
    // and folds the post-barrier LDS loads to undef (observed as ds=0 last round).
    #pragma unroll
    for (int i = tid; i < 256 * 5; i += 256) {
        if (i < nflt) {
            uint32_t lds_off = (uint32_t)(i * 4);   // byte offset of sro[i] in LDS
            const float* g = gsrc + i;
            asm volatile("global_load_async_to_lds_b32 %0, %1, off"
                         :: "v"(lds_off), "v"(g), "v"(&sro[i]) : "memory");
        }
    }
    asm volatile("s_wait_asynccnt 0" ::: "memory");
    __syncthreads();

    const int k = base + tid;
    if (k >= K) return;

    const float bidx = sro[tid * 5 + 0];
    const float x1   = sro[tid * 5 + 1];
    const float y1   = sro[tid * 5 + 2];
    const float x2   = sro[tid * 5 + 3];
    const float y2   = sro[tid * 5 + 4];

    // map_roi_levels: lvl = clip(floor(log2(sqrt(area)/56 + 1e-6)), 0, 3)
    const float scale = sqrtf(fmaxf((x2 - x1) * (y2 - y1), 0.0f));
    int lvl = (int)floorf(log2f(scale * (1.0f / 56.0f) + 1e-6f));
    lvl = max(0, min(3, lvl));

    const float ss = 1.0f / (float)(4 << lvl);   // spatial_scale = 1/stride

    RoiRec r;
    r.x1  = x1 * ss - 0.5f;                      // aligned=True offset
    r.y1  = y1 * ss - 0.5f;
    r.bw  = (x2 - x1) * ss * (1.0f / (float)OUTS);
    r.bh  = (y2 - y1) * ss * (1.0f / (float)OUTS);
    r.lvl = lvl;
    r.b   = (int)bidx;
    r.pad0 = 0; r.pad1 = 0;
    recs[k] = r;
}

// ---------------------------------------------------------------------------
// Kernel 2: one thread per output element, index = ((k*C + c)*49 + ph*7 + pw)
// so stores are fully coalesced with the (K,C,7,7) output layout.
// ---------------------------------------------------------------------------
__global__ __launch_bounds__(256)
void roi_align_kernel(const float* __restrict__ f0, const float* __restrict__ f1,
                      const float* __restrict__ f2, const float* __restrict__ f3,
                      const RoiRec* __restrict__ recs,
                      float* __restrict__ out, unsigned total)
{
    const unsigned idx = blockIdx.x * 256u + threadIdx.x;
    if (idx >= total) return;

    const int bin = (int)(idx % NBINS);
    const int pw  = bin % OUTS;
    const int ph  = bin / OUTS;
    const int c   = (int)((idx / NBINS) % CCH);
    const int k   = (int)(idx / (NBINS * CCH));

    const RoiRec r = recs[k];

    const float* fp; int H, W;
    switch (r.lvl) {                 // uniform per-RoI -> scalar branches
        case 0:  fp = f0; H = 200; W = 304; break;
        case 1:  fp = f1; H = 100; W = 152; break;
        case 2:  fp = f2; H = 50;  W = 76;  break;
        default: fp = f3; H = 25;  W = 38;  break;
    }
    const float Hf = (float)H, Wf = (float)W;
    const float* rbase = fp + (size_t)(r.b * CCH + c) * (size_t)(H * W);

    // Per-axis bilinear setup (torch/mmcv RoIAlign semantics), hoisted out of
    // the 2x2 sample loop: sample coord = origin + step*(bin + 0.25 + 0.5*s).
    int   x0a[SRK], x1a[SRK], y0a[SRK], y1a[SRK];
    float fxa[SRK], fya[SRK], vx[SRK], vy[SRK];

    #pragma unroll
    for (int s = 0; s < SRK; ++s) {
        float rx = r.x1 + r.bw * ((float)pw + 0.25f + 0.5f * (float)s);
        vx[s] = (rx >= -1.0f && rx <= Wf) ? 1.0f : 0.0f;
        float cx = fminf(fmaxf(rx, 0.0f), Wf - 1.0f);
        int lo = (int)cx;                        // cx >= 0 -> trunc == floor
        x0a[s] = lo;
        x1a[s] = min(lo + 1, W - 1);
        fxa[s] = cx - (float)lo;

        float ry = r.y1 + r.bh * ((float)ph + 0.25f + 0.5f * (float)s);
        vy[s] = (ry >= -1.0f && ry <= Hf) ? 1.0f : 0.0f;
        float cy = fminf(fmaxf(ry, 0.0f), Hf - 1.0f);
        int lyo = (int)cy;
        y0a[s] = lyo;
        y1a[s] = min(lyo + 1, H - 1);
        fya[s] = cy - (float)lyo;
    }

    // Prefetch the farthest-future gather line (gfx1250 global_prefetch_b8).
    __builtin_prefetch(rbase + (size_t)y1a[1] * W + x0a[0], 0, 0);

    float acc = 0.0f;
    #pragma unroll
    for (int sy = 0; sy < SRK; ++sy) {
        const float* row0 = rbase + (size_t)y0a[sy] * W;
        const float* row1 = rbase + (size_t)y1a[sy] * W;
        const float fy = fya[sy], gy = 1.0f - fy;
        #pragma unroll
        for (int sx = 0; sx < SRK; ++sx) {
            const float fx = fxa[sx], gx = 1.0f - fx;
            const float v00 = row0[x0a[sx]];
            const float v01 = row0[x1a[sx]];
            const float v10 = row1[x0a[sx]];
            const float v11 = row1[x1a[sx]];
            float val = (v00 * gx + v01 * fx) * gy + (v10 * gx + v11 * fx) * fy;
            acc = fmaf(val, vy[sy] * vx[sx], acc);   // mask invalid samples
        }
    }

    out[idx] = acc * 0.25f;   // mean over the 2x2 subsamples
}

// ---------------------------------------------------------------------------
extern "C" void kernel_launch(void* const* d_in, const int* in_sizes, int n_in,
                              void* d_out, int out_size, void* d_ws, size_t ws_size,
                              hipStream_t stream)
{
    const float* f0   = (const float*)d_in[0];   // (2,256,200,304)
    const float* f1   = (const float*)d_in[1];   // (2,256,100,152)
    const float* f2   = (const float*)d_in[2];   // (2,256,50,76)
    const float* f3   = (const float*)d_in[3];   // (2,256,25,38)
    const float* rois = (const float*)d_in[4];   // (K,5)

    const int K = in_sizes[4] / 5;
    RoiRec* recs = (RoiRec*)d_ws;

    const int sblocks = (K + 255) / 256;
    setup_rois_kernel<<<sblocks, 256, 0, stream>>>(rois, recs, K);

    const unsigned total = (unsigned)K * (unsigned)CCH * (unsigned)NBINS;
    const unsigned blocks = (total + 255u) / 256u;
    roi_align_kernel<<<blocks, 256, 0, stream>>>(f0, f1, f2, f3, recs,
                                                 (float*)d_out, total);
}